// GCN_47450798686231
// MI455X (gfx1250) — compile-verified
//
#include <hip/hip_runtime.h>

typedef __attribute__((ext_vector_type(2))) float v2f;
typedef __attribute__((ext_vector_type(8))) float v8f;

#define N_NODES 100000
#define N_EDGES 1600000

// ---------------- degree / symmetric norm ----------------
__global__ void k_init_deg(float* deg) {
    int i = blockIdx.x * blockDim.x + threadIdx.x;
    if (i < N_NODES) deg[i] = 1.0f;   // self-loop contribution
}

__global__ void k_count_deg(const long long* __restrict__ dst, float* deg) {
    int i = blockIdx.x * blockDim.x + threadIdx.x;
    if (i < N_EDGES) atomicAdd(&deg[(int)dst[i]], 1.0f);
}

__global__ void k_dinv(float* deg) {
    int i = blockIdx.x * blockDim.x + threadIdx.x;
    if (i < N_NODES) deg[i] = rsqrtf(deg[i]);   // deg >= 1 always (self-loop)
}

__global__ void k_zero(float* p, int n) {
    int i = blockIdx.x * blockDim.x + threadIdx.x;
    if (i < n) p[i] = 0.0f;
}

// ---------------- fp32 WMMA GEMM: C[M x N] = A[M x 128] * B[128 x N] ----------------
// One wave computes one 16(M) x 16(N) tile via 32x V_WMMA_F32_16X16X4_F32.
// Block covers 16 rows x N cols with N/16 waves; A tile staged in LDS (padded).
template <int N>
__global__ __launch_bounds__((N / 16) * 32) void k_gemm(const float* __restrict__ A,
                                                        const float* __restrict__ B,
                                                        float* __restrict__ C) {
    constexpr int NT  = (N / 16) * 32;  // threads per block
    constexpr int LDT = 132;            // padded LDS row stride (floats) -> conflict-free
    __shared__ float tileA[16 * LDT];

    const int row0 = blockIdx.x * 16;

    // Cooperative, fully-coalesced load of the 16x128 fp32 A tile (float4 per thread).
    for (int i = threadIdx.x; i < 16 * 32; i += NT) {
        const int r  = i >> 5;
        const int c4 = i & 31;
        const float4 v = reinterpret_cast<const float4*>(A + (size_t)(row0 + r) * 128)[c4];
        *reinterpret_cast<float4*>(&tileA[r * LDT + c4 * 4]) = v;
    }
    __syncthreads();

    const int wave = threadIdx.x >> 5;
    const int lane = threadIdx.x & 31;
    const int lh   = lane >> 4;   // lane half: selects K pair {0,1} vs {2,3}
    const int l16  = lane & 15;
    const int n0   = wave * 16;

    v8f acc = {};
    #pragma unroll 4
    for (int k = 0; k < 128; k += 4) {
        const int ka = k + lh * 2;
        // A fragment (16x4, M=l16, K=ka..ka+1) from LDS
        v2f a;
        a.x = tileA[l16 * LDT + ka];
        a.y = tileA[l16 * LDT + ka + 1];
        // B fragment (4x16, K=ka..ka+1, N=n0+l16) from global (weights are L2-resident)
        v2f b;
        b.x = B[(size_t)ka * N + n0 + l16];
        b.y = B[(size_t)(ka + 1) * N + n0 + l16];
        acc = __builtin_amdgcn_wmma_f32_16x16x4_f32(false, a, false, b, (short)0, acc,
                                                    false, false);
    }

    // C/D layout: VGPR i -> M = i + lh*8, N = l16
    #pragma unroll
    for (int i = 0; i < 8; ++i) {
        C[(size_t)(row0 + i + lh * 8) * N + n0 + l16] = acc[i];
    }
}

// ---------------- edge gather/scatter: agg[dst] += norm * h[src] ----------------
// One wave per edge; vectorized gather + f32 global atomics (resolve in L2).
template <int C>
__global__ void k_scatter(const float* __restrict__ h, const long long* __restrict__ src,
                          const long long* __restrict__ dst, const float* __restrict__ dinv,
                          float* __restrict__ agg) {
    const int e    = (blockIdx.x * blockDim.x + threadIdx.x) >> 5;
    const int lane = threadIdx.x & 31;
    if (e >= N_EDGES) return;

    const int s = (int)src[e];
    const int d = (int)dst[e];
    const float nm = dinv[s] * dinv[d];

    if constexpr (C == 128) {
        const float4 v = reinterpret_cast<const float4*>(h + (size_t)s * 128)[lane];
        float* ap = agg + (size_t)d * 128 + lane * 4;
        atomicAdd(ap + 0, v.x * nm);
        atomicAdd(ap + 1, v.y * nm);
        atomicAdd(ap + 2, v.z * nm);
        atomicAdd(ap + 3, v.w * nm);
    } else {
        const float2 v = reinterpret_cast<const float2*>(h + (size_t)s * 64)[lane];
        float* ap = agg + (size_t)d * 64 + lane * 2;
        atomicAdd(ap + 0, v.x * nm);
        atomicAdd(ap + 1, v.y * nm);
    }
}

// ---------------- add self-loop term + bias (+ optional relu), in place ----------------
template <int C, bool RELU>
__global__ void k_finalize(float* __restrict__ agg, const float* __restrict__ h,
                           const float* __restrict__ dinv, const float* __restrict__ bias) {
    const int idx = blockIdx.x * blockDim.x + threadIdx.x;
    if (idx >= N_NODES * C) return;
    const int node = idx / C;
    const int c    = idx & (C - 1);
    const float di = dinv[node];
    float v = agg[idx] + h[idx] * di * di + bias[c];
    if (RELU) v = fmaxf(v, 0.0f);
    agg[idx] = v;
}

extern "C" void kernel_launch(void* const* d_in, const int* in_sizes, int n_in,
                              void* d_out, int out_size, void* d_ws, size_t ws_size,
                              hipStream_t stream) {
    const float*     x   = (const float*)d_in[0];
    const long long* ei  = (const long long*)d_in[1];   // [2, N_EDGES] int64
    const float*     W1  = (const float*)d_in[2];
    const float*     b1  = (const float*)d_in[3];
    const float*     W2  = (const float*)d_in[4];
    const float*     b2  = (const float*)d_in[5];
    float*           out = (float*)d_out;

    const long long* src = ei;             // edge_index[0]
    const long long* dst = ei + N_EDGES;   // edge_index[1]

    char* ws = (char*)d_ws;
    float* dinv = (float*)ws;                                        // 100000 f32
    float* bufA = (float*)(ws + (1 << 19));                          // 100000*128 f32
    float* bufB = (float*)(ws + (1 << 19) + (size_t)N_NODES * 128 * 4);

    const int NB_NODE  = (N_NODES + 255) / 256;
    const int NB_EDGE  = (N_EDGES + 255) / 256;
    const int NB_EDGEW = (N_EDGES * 32 + 255) / 256;   // one wave per edge
    const int NB_128   = (N_NODES * 128 + 255) / 256;
    const int NB_64    = (N_NODES * 64 + 255) / 256;

    // normalization
    k_init_deg<<<NB_NODE, 256, 0, stream>>>(dinv);
    k_count_deg<<<NB_EDGE, 256, 0, stream>>>(dst, dinv);
    k_dinv<<<NB_NODE, 256, 0, stream>>>(dinv);

    // ---- layer 1: h = relu(aggregate(x @ W1) + b1) ----
    k_gemm<128><<<N_NODES / 16, 256, 0, stream>>>(x, W1, bufA);
    k_zero<<<NB_128, 256, 0, stream>>>(bufB, N_NODES * 128);
    k_scatter<128><<<NB_EDGEW, 256, 0, stream>>>(bufA, src, dst, dinv, bufB);
    k_finalize<128, true><<<NB_128, 256, 0, stream>>>(bufB, bufA, dinv, b1);

    // ---- layer 2: out = aggregate(h @ W2) + b2 ----
    k_gemm<64><<<N_NODES / 16, 128, 0, stream>>>(bufB, W2, bufA);   // h2 -> bufA
    k_zero<<<NB_64, 256, 0, stream>>>(out, N_NODES * 64);
    k_scatter<64><<<NB_EDGEW, 256, 0, stream>>>(bufA, src, dst, dinv, out);
    k_finalize<64, false><<<NB_64, 256, 0, stream>>>(out, bufA, dinv, b2);
}